// DecoderRNN_55448027792087
// MI455X (gfx1250) — compile-verified
//
#include <hip/hip_runtime.h>

#define H_ 256
#define V_ 50257
#define B_ 128
#define S_ 2048

typedef float v2f __attribute__((ext_vector_type(2)));
typedef float v8f __attribute__((ext_vector_type(8)));
typedef unsigned int u32x4 __attribute__((ext_vector_type(4)));
typedef int i32x4 __attribute__((ext_vector_type(4)));
typedef int i32x8 __attribute__((ext_vector_type(8)));

#if __has_builtin(__builtin_amdgcn_tensor_load_to_lds) && __has_builtin(__builtin_amdgcn_s_wait_tensorcnt)
#define USE_TDM 1
#else
#define USE_TDM 0
#endif

// ---------------- embedding gather ----------------
__global__ void embed_kernel(const int* __restrict__ seq,
                             const float* __restrict__ table,
                             float* __restrict__ out) {
    int idx = blockIdx.x * blockDim.x + threadIdx.x;   // 0 .. B*H-1
    int b = idx >> 8;
    int h = idx & 255;
    out[idx] = table[(size_t)seq[b] * H_ + h];
}

// ---------------- TDM tile load: 128 rows x 32 f32, row stride = stride_elems ----
// D# pad: 1 DWORD after every 32 DWORDs -> LDS row stride 33 (bank-conflict free).
// rows beyond rows_remaining are hardware zero-filled (tensor_dim1 OOB).
#define KTILE 32
#define LDSS  33
#define CHUNK_FLOATS (128 * LDSS)

__device__ __forceinline__ void tdm_load_tile(const float* gsrc,
                                              unsigned lds_byte_addr,
                                              unsigned rows_remaining,
                                              long long stride_elems) {
#if USE_TDM
    unsigned long long ga = (unsigned long long)(size_t)gsrc;
    u32x4 g0;
    g0[0] = 1u;                                  // count = 1 (valid user descriptor)
    g0[1] = lds_byte_addr;                       // D#.lds_addr
    g0[2] = (unsigned)(ga & 0xffffffffull);      // D#.global_addr[31:0]
    g0[3] = (unsigned)(ga >> 32) | (2u << 30);   // addr[56:32] | type=2 (image)
    unsigned td0 = 0x7fffffffu;                  // dim0 extent: K chunks always in-bounds
    unsigned td1 = rows_remaining;               // dim1 extent: OOB rows read as zero
    unsigned long long s0 = (unsigned long long)stride_elems;
    i32x8 g1;
    g1[0] = (int)((2u << 16)      // data_size = 4B
                | (1u << 20)      // pad_enable
                | (4u << 22));    // pad_interval: 32 DWORDs; pad_amount=0 -> 1 DWORD
    g1[1] = (int)((td0 & 0xffffu) << 16);                      // tensor_dim0 lo16
    g1[2] = (int)((td0 >> 16) | ((td1 & 0xffffu) << 16));      // dim0 hi16 | dim1 lo16
    g1[3] = (int)((td1 >> 16) | ((unsigned)KTILE << 16));      // dim1 hi16 | tile_dim0=32
    g1[4] = 128;                                               // tile_dim1=128, tile_dim2=0
    g1[5] = (int)(unsigned)(s0 & 0xffffffffull);               // dim0_stride lo32
    g1[6] = (int)(unsigned)(s0 >> 32);                         // dim0_stride hi16
    g1[7] = 0;
    i32x4 gz = {0, 0, 0, 0};
#if defined(__clang_major__) && (__clang_major__ >= 23)
    i32x8 gz8 = {0, 0, 0, 0, 0, 0, 0, 0};
    __builtin_amdgcn_tensor_load_to_lds(g0, g1, gz, gz, gz8, 0);
#else
    __builtin_amdgcn_tensor_load_to_lds(g0, g1, gz, gz, 0);
#endif
#else
    (void)gsrc; (void)lds_byte_addr; (void)rows_remaining; (void)stride_elems;
#endif
}

// ---------------- generic fp32 WMMA GEMM ----------------
// C[128, N] = A[128, K] * W[N, K]^T + bias ; act==1 -> tanh
// grid.x = ceil(N/128), block = 256 (8 waves). Wave w owns N-subtile w.
// W tiles double-buffered through the Tensor Data Mover; A tiles single-buffered.
__global__ void gemm_f32_wmma(const float* __restrict__ A, int lda,
                              const float* __restrict__ W, int ldw,
                              const float* __restrict__ bias,
                              float* __restrict__ C, int ldc,
                              int N, int K, int act) {
    __shared__ float As[CHUNK_FLOATS];
    __shared__ float Ws[2 * CHUNK_FLOATS];

    const int t    = threadIdx.x;
    const int wv   = t >> 5;
    const int lane = t & 31;
    const int n0   = blockIdx.x * 128;
    const int nchunks = K / KTILE;

    v8f acc[8];
    #pragma unroll
    for (int i = 0; i < 8; ++i)
        #pragma unroll
        for (int j = 0; j < 8; ++j) acc[i][j] = 0.0f;

#if USE_TDM
    const unsigned as_lds = (unsigned)(size_t)(void*)As;
    const unsigned ws_lds = (unsigned)(size_t)(void*)Ws;
    if (wv == 0)  // prologue: start W chunk 0 DMA into buffer 0
        tdm_load_tile(W + (size_t)n0 * ldw, ws_lds, (unsigned)(N - n0), ldw);
#endif

    for (int c = 0; c < nchunks; ++c) {
        const int kc = c * KTILE;
#if USE_TDM
        if (wv == 0) {
            tdm_load_tile(A + kc, as_lds, 128u, lda);                 // A chunk c
            if (c + 1 < nchunks)                                      // W chunk c+1
                tdm_load_tile(W + (size_t)n0 * ldw + (kc + KTILE),
                              ws_lds + (((c + 1) & 1) ? (unsigned)(CHUNK_FLOATS * 4) : 0u),
                              (unsigned)(N - n0), ldw);
        }
        // outstanding (wave 0, in order): W(c), A(c), [W(c+1)] -> keep W(c+1) in flight
        if (c + 1 < nchunks) __builtin_amdgcn_s_wait_tensorcnt(1);
        else                 __builtin_amdgcn_s_wait_tensorcnt(0);
        __syncthreads();
        const float* Wb = Ws + ((c & 1) ? CHUNK_FLOATS : 0);
#else
        // fallback: cooperative vector loads
        #pragma unroll
        for (int r = 0; r < 16; ++r) {
            int idx = t + 256 * r;            // 0..4095
            int row = idx >> 5;
            int col = idx & 31;
            As[row * LDSS + col] = A[(size_t)row * lda + kc + col];
            int grow = n0 + row;
            Ws[row * LDSS + col] =
                (grow < N) ? W[(size_t)grow * ldw + kc + col] : 0.0f;
        }
        if (kc + KTILE < K) {
            int row = t >> 5, col = t & 31;
            int grow = n0 + row;
            if (grow < N)
                __builtin_prefetch(&W[(size_t)grow * ldw + kc + KTILE + col], 0, 1);
        }
        __syncthreads();
        const float* Wb = Ws;
#endif
        // compute: batch 1 B-frag + 8 A-frag LDS loads, then 8 back-to-back WMMAs
        const int rowa  = lane & 15;
        const int khalf = (lane >> 4) << 1;       // K interleave across lane halves
        #pragma unroll
        for (int k0 = 0; k0 < KTILE; k0 += 4) {
            v2f bfrag;
            {
                const float* bp = &Wb[(wv * 16 + rowa) * LDSS + k0 + khalf];
                bfrag.x = bp[0];
                bfrag.y = bp[1];
            }
            v2f af[8];
            #pragma unroll
            for (int mt = 0; mt < 8; ++mt) {
                const float* ap = &As[((mt << 4) + rowa) * LDSS + k0 + khalf];
                af[mt].x = ap[0];
                af[mt].y = ap[1];
            }
            #pragma unroll
            for (int mt = 0; mt < 8; ++mt)
                acc[mt] = __builtin_amdgcn_wmma_f32_16x16x4_f32(
                    false, af[mt], false, bfrag, (short)0, acc[mt], false, false);
        }
        __syncthreads();
    }

    // epilogue: C/D layout -> VGPR i holds row i (lanes 0-15) / row i+8 (lanes 16-31)
    int nG = n0 + wv * 16 + (lane & 15);
    if (nG < N) {
        float bv = bias[nG];
        int mhalf = (lane >> 4) << 3;             // 0 or 8
        #pragma unroll
        for (int mt = 0; mt < 8; ++mt)
            #pragma unroll
            for (int i = 0; i < 8; ++i) {
                int m = (mt << 4) + i + mhalf;
                float v = acc[mt][i] + bv;
                if (act == 1) v = tanhf(v);
                C[(size_t)m * ldc + nG] = v;
            }
    }
}

// ---------------- GRU gate math ----------------
__global__ void gru_gates_kernel(const float* __restrict__ gi,
                                 const float* __restrict__ gh,
                                 const float* __restrict__ hprev,
                                 float* __restrict__ hout,
                                 float* __restrict__ hid_out,
                                 float* __restrict__ concat_dst) {
    int idx = blockIdx.x * blockDim.x + threadIdx.x;  // 0..B*H-1
    int b = idx >> 8;
    int j = idx & 255;
    int g = b * (3 * H_) + j;
    float r = 1.0f / (1.0f + __expf(-(gi[g] + gh[g])));
    float z = 1.0f / (1.0f + __expf(-(gi[g + H_] + gh[g + H_])));
    float n = tanhf(gi[g + 2 * H_] + r * gh[g + 2 * H_]);
    float h = (1.0f - z) * n + z * hprev[idx];
    hout[idx] = h;
    hid_out[idx] = h;
    if (concat_dst) concat_dst[b * (2 * H_) + j] = h;
}

// ---------------- dot attention + softmax + context ----------------
__global__ void attention_kernel(const float* __restrict__ h1,
                                 const float* __restrict__ enc,
                                 float* __restrict__ attn_out,
                                 float* __restrict__ concat) {
    __shared__ float h1s[H_];
    __shared__ float Es[S_];
    __shared__ float red[16];

    const int b = blockIdx.x;
    const int t = threadIdx.x;
    const int wv = t >> 5;
    const int lane = t & 31;

    h1s[t] = h1[b * H_ + t];
    __syncthreads();

    // pass 1: energies e_s = dot(h1[b], enc[s,b,:]) ; one s per wave
    for (int s = wv; s < S_; s += 8) {
        const float* p = enc + ((size_t)s * B_ + b) * H_ + lane * 8;
        const float* q = h1s + lane * 8;
        float part = 0.0f;
        #pragma unroll
        for (int j = 0; j < 8; ++j) part += p[j] * q[j];
        #pragma unroll
        for (int off = 16; off > 0; off >>= 1)
            part += __shfl_xor(part, off, 32);
        if (lane == 0) Es[s] = part;
    }
    __syncthreads();

    // softmax max
    float lm = -3.4e38f;
    for (int i = t; i < S_; i += 256) lm = fmaxf(lm, Es[i]);
    #pragma unroll
    for (int off = 16; off > 0; off >>= 1)
        lm = fmaxf(lm, __shfl_xor(lm, off, 32));
    if (lane == 0) red[wv] = lm;
    __syncthreads();
    float m = fmaxf(fmaxf(fmaxf(red[0], red[1]), fmaxf(red[2], red[3])),
                    fmaxf(fmaxf(red[4], red[5]), fmaxf(red[6], red[7])));

    // exp + sum
    float ls = 0.0f;
    for (int i = t; i < S_; i += 256) {
        float e = __expf(Es[i] - m);
        Es[i] = e;
        ls += e;
    }
    #pragma unroll
    for (int off = 16; off > 0; off >>= 1) ls += __shfl_xor(ls, off, 32);
    if (lane == 0) red[8 + wv] = ls;
    __syncthreads();
    float tot = (red[8] + red[9]) + (red[10] + red[11]) +
                (red[12] + red[13]) + (red[14] + red[15]);
    float inv = 1.0f / tot;

    for (int i = t; i < S_; i += 256) {
        float w = Es[i] * inv;
        Es[i] = w;
        attn_out[(size_t)b * S_ + i] = w;
    }
    __syncthreads();

    // pass 2: context[h] = sum_s w_s * enc[s,b,h] ; thread t owns h = t
    float a0 = 0.f, a1 = 0.f, a2 = 0.f, a3 = 0.f;
    const float* base = enc + (size_t)b * H_ + t;
    const size_t stride = (size_t)B_ * H_;
    for (int s = 0; s < S_; s += 4) {
        a0 += Es[s + 0] * base[(size_t)(s + 0) * stride];
        a1 += Es[s + 1] * base[(size_t)(s + 1) * stride];
        a2 += Es[s + 2] * base[(size_t)(s + 2) * stride];
        a3 += Es[s + 3] * base[(size_t)(s + 3) * stride];
    }
    concat[b * (2 * H_) + H_ + t] = (a0 + a1) + (a2 + a3);
}

// ---------------- host launcher ----------------
extern "C" void kernel_launch(void* const* d_in, const int* in_sizes, int n_in,
                              void* d_out, int out_size, void* d_ws, size_t ws_size,
                              hipStream_t stream) {
    (void)in_sizes; (void)n_in; (void)out_size; (void)ws_size;
    const int*   seq   = (const int*)  d_in[0];
    const float* lastH = (const float*)d_in[1];   // [2,B,H]
    const float* enc   = (const float*)d_in[2];   // [S,B,H]
    const float* embT  = (const float*)d_in[3];   // [V,H]
    const float* w_ih0 = (const float*)d_in[4];
    const float* w_hh0 = (const float*)d_in[5];
    const float* b_ih0 = (const float*)d_in[6];
    const float* b_hh0 = (const float*)d_in[7];
    const float* w_ih1 = (const float*)d_in[8];
    const float* w_hh1 = (const float*)d_in[9];
    const float* b_ih1 = (const float*)d_in[10];
    const float* b_hh1 = (const float*)d_in[11];
    const float* cw    = (const float*)d_in[12];  // [H,2H]
    const float* cb    = (const float*)d_in[13];
    const float* ow    = (const float*)d_in[14];  // [V,H]
    const float* ob    = (const float*)d_in[15];

    float* ws    = (float*)d_ws;
    float* emb   = ws;             // 32768
    float* gi    = ws + 32768;     // 98304
    float* gh    = ws + 131072;    // 98304
    float* h0    = ws + 229376;    // 32768
    float* h1    = ws + 262144;    // 32768
    float* cat   = ws + 294912;    // 65536  [B, 2H]
    float* catac = ws + 360448;    // 32768

    float* out  = (float*)d_out;                  // [B,V]
    float* hid0 = out + (size_t)B_ * V_;          // [B,H]
    float* hid1 = hid0 + B_ * H_;                 // [B,H]
    float* attn = hid1 + B_ * H_;                 // [B,S]

    embed_kernel<<<(B_ * H_) / 256, 256, 0, stream>>>(seq, embT, emb);

    // GRU layer 0
    gemm_f32_wmma<<<6, 256, 0, stream>>>(emb,   H_, w_ih0, H_, b_ih0, gi, 3 * H_, 3 * H_, H_, 0);
    gemm_f32_wmma<<<6, 256, 0, stream>>>(lastH, H_, w_hh0, H_, b_hh0, gh, 3 * H_, 3 * H_, H_, 0);
    gru_gates_kernel<<<B_, 256, 0, stream>>>(gi, gh, lastH, h0, hid0, nullptr);

    // GRU layer 1 (also deposits h1 into concat[:, 0:H])
    gemm_f32_wmma<<<6, 256, 0, stream>>>(h0,              H_, w_ih1, H_, b_ih1, gi, 3 * H_, 3 * H_, H_, 0);
    gemm_f32_wmma<<<6, 256, 0, stream>>>(lastH + B_ * H_, H_, w_hh1, H_, b_hh1, gh, 3 * H_, 3 * H_, H_, 0);
    gru_gates_kernel<<<B_, 256, 0, stream>>>(gi, gh, lastH + B_ * H_, h1, hid1, cat);

    // attention: attn weights straight to d_out, context into concat[:, H:2H]
    attention_kernel<<<B_, 256, 0, stream>>>(h1, enc, attn, cat);

    // concat layer: tanh([h1, ctx] @ cw^T + cb)
    gemm_f32_wmma<<<2, 256, 0, stream>>>(cat, 2 * H_, cw, 2 * H_, cb, catac, H_, H_, 2 * H_, 1);

    // output projection: [B,V] = catac @ ow^T + ob
    gemm_f32_wmma<<<(V_ + 127) / 128, 256, 0, stream>>>(catac, H_, ow, H_, ob, out, V_, V_, H_, 0);
}